// DPAR_90615220011749
// MI455X (gfx1250) — compile-verified
//
#include <hip/hip_runtime.h>
#include <hip/hip_bf16.h>
#include <math.h>

#define N_NODES 100000
#define N_FEAT  512
#define HIDDEN  128
#define N_CLASS 16
#define N_EDGES 3200000
#define ALPHA   0.25f

typedef __attribute__((ext_vector_type(2))) float v2f;
typedef __attribute__((ext_vector_type(8))) float v8f;

// ---------------------------------------------------------------------------
// Fused GEMM kernel:  local = relu(x @ W1) @ W2
// One wave (32 lanes) owns a 16-row tile. 8 waves / block -> 128 rows / block.
// W1 is staged in LDS in 64-row K-chunks (cooperatively loaded by the block).
// GEMM1 accumulates 8 n-tiles of 16x16 f32 via V_WMMA_F32_16X16X4_F32.
// Relu'd hidden tile is staged in LDS to convert C/D layout -> A layout,
// then GEMM2 (16x128 @ 128x16) runs with W2 B-fragments read from global.
// ---------------------------------------------------------------------------
#define KCHUNK   64
#define W1C_STR  136   // 136*2 mod 64 == 16 -> B-frag reads conflict-free
#define HID_STR  132   // 132   mod 64 ==  4 -> A-frag reads conflict-free

__global__ __launch_bounds__(256) void gcn_gemm_kernel(
    const float* __restrict__ x, const float* __restrict__ W1,
    const float* __restrict__ W2, float* __restrict__ local_out)
{
    __shared__ float w1c[KCHUNK][W1C_STR];     //  ~34 KB
    __shared__ float hid[8][16][HID_STR];      //  ~66 KB

    const int tid  = threadIdx.x;
    const int wave = tid >> 5;
    const int lane = tid & 31;
    const int half = lane >> 4;   // 0 or 1
    const int l16  = lane & 15;

    const int row_tile = blockIdx.x * 8 + wave;
    const int row_base = row_tile * 16;
    const bool active  = (row_base < N_NODES);   // wave-uniform (N%16==0)

    v8f acc[8];
    for (int t = 0; t < 8; ++t)
        for (int i = 0; i < 8; ++i) acc[t][i] = 0.0f;

    for (int k0 = 0; k0 < N_FEAT; k0 += KCHUNK) {
        __syncthreads();   // protect previous chunk's readers
        // Cooperative load of W1[k0..k0+63][0..127] into LDS (float4 lanes).
        for (int i = tid; i < KCHUNK * (HIDDEN / 4); i += 256) {
            const int r  = i >> 5;          // HIDDEN/4 == 32 float4 per row
            const int c4 = i & 31;
            const float4 v = ((const float4*)(W1 + (size_t)(k0 + r) * HIDDEN))[c4];
            *(float4*)&w1c[r][c4 * 4] = v;
        }
        __syncthreads();

        if (active) {
            for (int kk = 0; kk < KCHUNK; kk += 4) {
                // A fragment: row M = l16, K = k0+kk + 2*half + {0,1}
                const float* ap = x + (size_t)(row_base + l16) * N_FEAT
                                    + (k0 + kk + 2 * half);
                v2f a = *(const v2f*)ap;          // 8B aligned
                for (int t = 0; t < 8; ++t) {
                    v2f b;                        // col N = t*16+l16
                    b[0] = w1c[kk + 2 * half + 0][t * 16 + l16];
                    b[1] = w1c[kk + 2 * half + 1][t * 16 + l16];
                    acc[t] = __builtin_amdgcn_wmma_f32_16x16x4_f32(
                        false, a, false, b, (short)0, acc[t], false, false);
                }
            }
        }
    }

    if (active) {
        // Relu + stage hidden tile: acc[t][r] is (M = r+8*half, N = t*16+l16).
        for (int t = 0; t < 8; ++t)
            for (int r = 0; r < 8; ++r) {
                float v = acc[t][r];
                hid[wave][r + 8 * half][t * 16 + l16] = v > 0.0f ? v : 0.0f;
            }
        // GEMM2: (16x128) @ W2(128x16)
        v8f c2;
        for (int i = 0; i < 8; ++i) c2[i] = 0.0f;
        for (int kk = 0; kk < HIDDEN; kk += 4) {
            v2f a, b;
            a[0] = hid[wave][l16][kk + 2 * half + 0];
            a[1] = hid[wave][l16][kk + 2 * half + 1];
            b[0] = W2[(size_t)(kk + 2 * half + 0) * N_CLASS + l16];
            b[1] = W2[(size_t)(kk + 2 * half + 1) * N_CLASS + l16];
            c2 = __builtin_amdgcn_wmma_f32_16x16x4_f32(
                false, a, false, b, (short)0, c2, false, false);
        }
        for (int r = 0; r < 8; ++r)
            local_out[(size_t)(row_base + r + 8 * half) * N_CLASS + l16] = c2[r];
    }
}

// ---------------------------------------------------------------------------
__global__ void zero_kernel(float* __restrict__ p, long long n)
{
    long long i = (long long)blockIdx.x * blockDim.x + threadIdx.x;
    if (i < n) p[i] = 0.0f;
}

__global__ void deg_kernel(const int* __restrict__ src, float* __restrict__ deg)
{
    int e = blockIdx.x * blockDim.x + threadIdx.x;
    if (e < N_EDGES) atomicAdd(&deg[src[e]], 1.0f);
}

// One thread per (edge, class): agg[src[e]*16+c] += logits[dst[e]*16+c]
__global__ void scatter_kernel(const int* __restrict__ src,
                               const int* __restrict__ dst,
                               const float* __restrict__ logits,
                               float* __restrict__ agg)
{
    long long gid = (long long)blockIdx.x * blockDim.x + threadIdx.x;
    if (gid >= (long long)N_EDGES * N_CLASS) return;
    const int e = (int)(gid >> 4);
    const int c = (int)(gid & 15);
    const int s = src[e];
    const int d = dst[e];
    atomicAdd(&agg[(size_t)s * N_CLASS + c], logits[(size_t)d * N_CLASS + c]);
}

__global__ void combine_kernel(const float* __restrict__ agg,
                               const float* __restrict__ local,
                               const float* __restrict__ deg,
                               float* __restrict__ out)
{
    long long i = (long long)blockIdx.x * blockDim.x + threadIdx.x;
    if (i >= (long long)N_NODES * N_CLASS) return;
    const int row = (int)(i >> 4);
    const float scale = (1.0f - ALPHA) / fmaxf(deg[row], 1e-12f);
    out[i] = scale * agg[i] + ALPHA * local[i];
}

// Final combine + row-wise log_softmax (16 classes, one thread per row).
__global__ void final_kernel(const float* __restrict__ agg,
                             const float* __restrict__ local,
                             const float* __restrict__ deg,
                             float* __restrict__ out)
{
    int row = blockIdx.x * blockDim.x + threadIdx.x;
    if (row >= N_NODES) return;
    const float scale = (1.0f - ALPHA) / fmaxf(deg[row], 1e-12f);
    float t[N_CLASS];
    float m = -INFINITY;
    for (int c = 0; c < N_CLASS; ++c) {
        t[c] = scale * agg[(size_t)row * N_CLASS + c]
             + ALPHA * local[(size_t)row * N_CLASS + c];
        m = fmaxf(m, t[c]);
    }
    float s = 0.0f;
    for (int c = 0; c < N_CLASS; ++c) s += __expf(t[c] - m);
    const float lse = m + __logf(s);
    for (int c = 0; c < N_CLASS; ++c)
        out[(size_t)row * N_CLASS + c] = t[c] - lse;
}

// ---------------------------------------------------------------------------
extern "C" void kernel_launch(void* const* d_in, const int* in_sizes, int n_in,
                              void* d_out, int out_size, void* d_ws, size_t ws_size,
                              hipStream_t stream)
{
    const float* x    = (const float*)d_in[0];
    const float* W1   = (const float*)d_in[1];
    const float* W2   = (const float*)d_in[2];
    const int*   esrc = (const int*)d_in[3];
    const int*   edst = (const int*)d_in[4];
    float* out = (float*)d_out;

    float* local  = (float*)d_ws;                         // N*16 f32
    float* logits = local  + (size_t)N_NODES * N_CLASS;   // N*16 f32
    float* deg    = logits + (size_t)N_NODES * N_CLASS;   // N    f32
    float* agg    = out;                                  // reuse d_out as agg

    const dim3 blk(256);
    const long long nl   = (long long)N_NODES * N_CLASS;
    const long long ne16 = (long long)N_EDGES * N_CLASS;
    const unsigned gN    = (unsigned)((N_NODES + 255) / 256);
    const unsigned gNL   = (unsigned)((nl + 255) / 256);
    const unsigned gE    = (unsigned)((N_EDGES + 255) / 256);
    const unsigned gE16  = (unsigned)((ne16 + 255) / 256);

    // 1) local = relu(x@W1)@W2  (6250 16-row tiles, 8 per block)
    const unsigned gGEMM = (unsigned)(((N_NODES + 15) / 16 + 7) / 8);
    gcn_gemm_kernel<<<gGEMM, blk, 0, stream>>>(x, W1, W2, local);

    // 2) degree
    zero_kernel<<<gN, blk, 0, stream>>>(deg, N_NODES);
    deg_kernel<<<gE, blk, 0, stream>>>(esrc, deg);

    // 3) iteration 1: agg = A @ local ; logits = scale*agg + alpha*local
    zero_kernel<<<gNL, blk, 0, stream>>>(agg, nl);
    scatter_kernel<<<gE16, blk, 0, stream>>>(esrc, edst, local, agg);
    combine_kernel<<<gNL, blk, 0, stream>>>(agg, local, deg, logits);

    // 4) iteration 2 + log_softmax
    zero_kernel<<<gNL, blk, 0, stream>>>(agg, nl);
    scatter_kernel<<<gE16, blk, 0, stream>>>(esrc, edst, logits, agg);
    final_kernel<<<gN, blk, 0, stream>>>(agg, local, deg, out);
}